// DenoiseNet_37709812859383
// MI455X (gfx1250) — compile-verified
//
#include <hip/hip_runtime.h>

typedef float v2f __attribute__((ext_vector_type(2)));
typedef float v8f __attribute__((ext_vector_type(8)));

#define B_    4
#define N_    10000
#define M_    10000
#define P_    128
#define KS_   32
#define Z_    128
#define SIGMA_ 0.01f

// D = A(16x4,f32) * B(4x16,f32) + C(16x16,f32)   -- exact f32 FMA semantics
static __device__ __forceinline__ v8f wmma_f32_k4(v2f a, v2f b, v8f c) {
  return __builtin_amdgcn_wmma_f32_16x16x4_f32(false, a, false, b, (short)0, c, false, false);
}

// Async memory->LDS copy (CDNA5): copies the float at gptr to the LDS byte offset lds_byte_off, tracked by ASYNCcnt.
static __device__ __forceinline__ void async_ld_f32(unsigned lds_byte_off, const float* gptr) {
  asm volatile("global_load_async_to_lds_b32 %0, %1, off"
               :: "v"(lds_byte_off), "v"(gptr) : "memory");
}
static __device__ __forceinline__ void wait_async0() {
  asm volatile("s_wait_asynccnt 0x0" ::: "memory");
}

// ---------------- Kernel A: gather sampled points + feature MLP (3->64 relu ->128) ----------
// grid 32 x block 32 (1 wave). Each wave owns 16 rows of the 512-row (b,p) space.
__global__ __launch_bounds__(32) void k_feat(
    const float* __restrict__ noisy_pc, const int* __restrict__ sampled_idx,
    const float* __restrict__ Wf1, const float* __restrict__ bf1,
    const float* __restrict__ Wf2, const float* __restrict__ bf2,
    float* __restrict__ f_origin, float* __restrict__ noisy_feat)
{
  __shared__ float h1[16][68];                    // 16 rows x 64 cols (padded stride)
  const int lane = threadIdx.x;
  const bool hi  = lane >= 16;
  const int  lc  = lane & 15;
  const int  tile = blockIdx.x;
  const int  row  = tile * 16 + lc;               // global (b*P+p) row
  const int  b = row >> 7, p = row & 127;
  const int  nidx = sampled_idx[p];
  const float* pt = noisy_pc + ((long)b * N_ + nidx) * 3;
  const float px = pt[0], py = pt[1], pz = pt[2];

  // A tile: K padded 3->4 with zeros. lanes<16: (K0,K1)=(x,y); lanes>=16: (K2,K3)=(z,0)
  v2f a; a[0] = hi ? pz : px; a[1] = hi ? 0.0f : py;
  if (!hi) { f_origin[row*3+0]=px; f_origin[row*3+1]=py; f_origin[row*3+2]=pz; }

  // layer 1: (16x4)x(4x64) -- unconditional loads, value-select via cndmask (no exec churn)
  for (int nt = 0; nt < 4; ++nt) {
    const int col = nt*16 + lc;
    const float w0 = Wf1[0*64 + col];
    const float w1 = Wf1[1*64 + col];
    const float w2 = Wf1[2*64 + col];
    v2f bm; bm[0] = hi ? w2 : w0; bm[1] = hi ? 0.0f : w1;
    v8f c = {};
    c = wmma_f32_k4(a, bm, c);
    const float bias = bf1[col];
    #pragma unroll
    for (int r = 0; r < 8; ++r)
      h1[r + (hi ? 8 : 0)][col] = fmaxf(c[r] + bias, 0.0f);
  }
  __syncthreads();
  // layer 2: (16x64)x(64x128), 16 K-chunks
  for (int nt = 0; nt < 8; ++nt) {
    const int col = nt*16 + lc;
    v8f c = {};
    for (int kk = 0; kk < 16; ++kk) {
      const int kb = kk*4;
      const int r0 = kb + (hi?2:0), r1 = kb + (hi?3:1);
      v2f a2, b2;
      a2[0] = h1[lc][r0];            a2[1] = h1[lc][r1];
      b2[0] = Wf2[r0*Z_ + col];      b2[1] = Wf2[r1*Z_ + col];
      c = wmma_f32_k4(a2, b2, c);
    }
    const float bias = bf2[col];
    #pragma unroll
    for (int r = 0; r < 8; ++r)
      noisy_feat[(tile*16 + r + (hi?8:0))*Z_ + col] = c[r] + bias;
  }
}

// ---------------- Kernel B: KNN-32 of 512 sampled points vs noisy cloud + gather -------------
// grid B x block 128; one query per thread; noisy tiles streamed to LDS via async loads.
__global__ __launch_bounds__(128) void k_knn1(
    const float* __restrict__ noisy_pc, const float* __restrict__ f_origin,
    float* __restrict__ fgat)
{
  __shared__ float pts[128*3];
  __shared__ float topd[128][KS_];
  __shared__ int   topi[128][KS_];
  const int tid = threadIdx.x;
  const int b   = blockIdx.x;
  const int q   = b*P_ + tid;
  const float qx = f_origin[q*3+0], qy = f_origin[q*3+1], qz = f_origin[q*3+2];
  for (int k = 0; k < KS_; ++k) { topd[tid][k] = 3.0e38f; topi[tid][k] = 0; }
  float worst = 3.0e38f; int wslot = 0;
  const float* base = noisy_pc + (long)b * N_ * 3;
  for (int j0 = 0; j0 < N_; j0 += 128) {
    const int cnt = min(128, N_ - j0);
    __syncthreads();
    for (int t = tid; t < cnt*3; t += 128)
      async_ld_f32((unsigned)(unsigned long long)&pts[t], base + j0*3 + t);
    wait_async0();
    __syncthreads();
    for (int j = 0; j < cnt; ++j) {
      const float dx = qx - pts[j*3+0];
      const float dy = qy - pts[j*3+1];
      const float dz = qz - pts[j*3+2];
      const float d2 = dx*dx + dy*dy + dz*dz;
      if (d2 < worst) {                      // replace current worst, rescan for new worst
        topd[tid][wslot] = d2; topi[tid][wslot] = j0 + j;
        worst = -1.0f;
        for (int s = 0; s < KS_; ++s) {
          const float v = topd[tid][s];
          if (v > worst) { worst = v; wslot = s; }
        }
      }
    }
  }
  for (int k = 0; k < KS_; ++k) {            // gather neighbor coords (order irrelevant to loss)
    const float* nb = base + (long)topi[tid][k]*3;
    float* o = fgat + ((long)q*KS_ + k)*3;
    o[0]=nb[0]; o[1]=nb[1]; o[2]=nb[2];
  }
}

// ---------------- Kernel C: score MLP 131->128 relu ->64 relu ->3 via WMMA -------------------
// grid 512 (one per (b,p)) x block 64 (2 waves; wave w owns k-rows w*16..w*16+15).
__global__ __launch_bounds__(64) void k_score(
    const float* __restrict__ fgat, const float* __restrict__ f_origin,
    const float* __restrict__ noisy_feat,
    const float* __restrict__ Ws1, const float* __restrict__ bs1,
    const float* __restrict__ Ws2, const float* __restrict__ bs2,
    const float* __restrict__ Ws3, const float* __restrict__ bs3,
    float* __restrict__ estim)
{
  __shared__ float hin[32][132];   // [x(3)|z(128)|pad]; stride%64==4 -> conflict-free row reads
  __shared__ float h1s[32][132];
  __shared__ float h2s[32][68];
  __shared__ float w3s[64][16];    // Ws3 zero-padded 64x3 -> 64x16 (kills divergent loads)
  const int tid = threadIdx.x;
  const int bp  = blockIdx.x;
  for (int t = tid; t < 32*132; t += 64) {
    const int r = t / 132, c = t % 132;
    float v;
    if (c < 3)        v = fgat[((long)bp*KS_ + r)*3 + c] - f_origin[bp*3 + c];
    else if (c < 131) v = noisy_feat[bp*Z_ + (c-3)];
    else              v = 0.0f;
    hin[r][c] = v;
  }
  for (int t = tid; t < 64*16; t += 64) {
    const int r = t >> 4, c = t & 15;
    w3s[r][c] = (c < 3) ? Ws3[r*3 + c] : 0.0f;
  }
  __syncthreads();
  const int wave = tid >> 5, lane = tid & 31;
  const bool hi = lane >= 16;
  const int lc = lane & 15;
  const int ar = wave*16 + lc;
  // layer 1: K=131 (pad 132, 33 chunks), N=128
  // bm[0] row is always <=130; bm[1] row can be 131 only in the last chunk:
  // clamp the address (always in-bounds) and cndmask the value -> branch-free.
  for (int nt = 0; nt < 8; ++nt) {
    const int col = nt*16 + lc;
    v8f c = {};
    for (int kk = 0; kk < 33; ++kk) {
      const int kb = kk*4;
      const int r0 = kb + (hi?2:0), r1 = kb + (hi?3:1);
      const int r1c = min(r1, 130);
      v2f a, bm;
      a[0] = hin[ar][r0]; a[1] = hin[ar][r1];
      bm[0] = Ws1[r0*128 + col];
      const float w1v = Ws1[r1c*128 + col];
      bm[1] = (r1 < 131) ? w1v : 0.0f;
      c = wmma_f32_k4(a, bm, c);
    }
    const float bias = bs1[col];
    #pragma unroll
    for (int r = 0; r < 8; ++r)
      h1s[wave*16 + r + (hi?8:0)][col] = fmaxf(c[r] + bias, 0.0f);
  }
  __syncthreads();
  // layer 2: K=128 (32 chunks), N=64
  for (int nt = 0; nt < 4; ++nt) {
    const int col = nt*16 + lc;
    v8f c = {};
    for (int kk = 0; kk < 32; ++kk) {
      const int kb = kk*4;
      const int r0 = kb + (hi?2:0), r1 = kb + (hi?3:1);
      v2f a, bm;
      a[0] = h1s[ar][r0];       a[1] = h1s[ar][r1];
      bm[0] = Ws2[r0*64 + col]; bm[1] = Ws2[r1*64 + col];
      c = wmma_f32_k4(a, bm, c);
    }
    const float bias = bs2[col];
    #pragma unroll
    for (int r = 0; r < 8; ++r)
      h2s[wave*16 + r + (hi?8:0)][col] = fmaxf(c[r] + bias, 0.0f);
  }
  __syncthreads();
  // layer 3: K=64 (16 chunks), N=3 padded to 16; B from zero-padded LDS tile
  {
    const int col = lc;
    v8f c = {};
    for (int kk = 0; kk < 16; ++kk) {
      const int kb = kk*4;
      const int r0 = kb + (hi?2:0), r1 = kb + (hi?3:1);
      v2f a, bm;
      a[0] = h2s[ar][r0]; a[1] = h2s[ar][r1];
      bm[0] = w3s[r0][col]; bm[1] = w3s[r1][col];
      c = wmma_f32_k4(a, bm, c);
    }
    if (col < 3) {
      const float bias = bs3[col];
      #pragma unroll
      for (int r = 0; r < 8; ++r) {
        const int m = wave*16 + r + (hi?8:0);
        estim[((long)bp*KS_ + m)*3 + col] = c[r] + bias;
      }
    }
  }
}

// ---------------- Kernel D: KNN-4 vs clean cloud + ground score (dominant work) --------------
// grid 64 x block 256; one query/thread; register-resident sorted top-4; async LDS staging.
__global__ __launch_bounds__(256) void k_knn2(
    const float* __restrict__ clean_pc, const float* __restrict__ fgat,
    float* __restrict__ ground)
{
  __shared__ float pts[256*3];
  const int tid = threadIdx.x;
  const long qid = (long)blockIdx.x * 256 + tid;   // 4096 queries per batch -> block is 1 batch
  const int b = (int)(qid >> 12);
  const float qx = fgat[qid*3+0], qy = fgat[qid*3+1], qz = fgat[qid*3+2];
  float d0=3e38f,d1=3e38f,d2v=3e38f,d3=3e38f;
  int   i0=0,i1=0,i2=0,i3=0;
  const float* base = clean_pc + (long)b * M_ * 3;
  for (int j0 = 0; j0 < M_; j0 += 256) {
    const int cnt = min(256, M_ - j0);
    __syncthreads();
    for (int t = tid; t < cnt*3; t += 256)
      async_ld_f32((unsigned)(unsigned long long)&pts[t], base + j0*3 + t);
    wait_async0();
    __syncthreads();
    for (int j = 0; j < cnt; ++j) {
      const float dx = qx - pts[j*3+0];
      const float dy = qy - pts[j*3+1];
      const float dz = qz - pts[j*3+2];
      const float d = dx*dx + dy*dy + dz*dz;
      if (d < d3) {
        const int jj = j0 + j;
        if      (d < d0)  { d3=d2v;i3=i2; d2v=d1;i2=i1; d1=d0;i1=i0; d0=d;i0=jj; }
        else if (d < d1)  { d3=d2v;i3=i2; d2v=d1;i2=i1; d1=d;i1=jj; }
        else if (d < d2v) { d3=d2v;i3=i2; d2v=d;i2=jj; }
        else              { d3=d; i3=jj; }
      }
    }
  }
  float gx=0.f, gy=0.f, gz=0.f;
  const int ids[4] = {i0,i1,i2,i3};
  #pragma unroll
  for (int k = 0; k < 4; ++k) {
    const float* nb = base + (long)ids[k]*3;
    gx += nb[0]; gy += nb[1]; gz += nb[2];
  }
  ground[qid*3+0] = gx*0.25f - qx;   // mean(nn - f) == mean(nn) - f
  ground[qid*3+1] = gy*0.25f - qy;
  ground[qid*3+2] = gz*0.25f - qz;
}

// ---------------- Kernel E: scalar loss reduction --------------------------------------------
__global__ __launch_bounds__(256) void k_loss(
    const float* __restrict__ estim, const float* __restrict__ ground,
    float* __restrict__ out)
{
  __shared__ float red[256];
  const int tid = threadIdx.x;
  float acc = 0.0f;
  for (int i = tid; i < B_*P_*KS_*3; i += 256) {
    const float df = estim[i] - ground[i];
    acc += df*df;
  }
  red[tid] = acc;
  __syncthreads();
  for (int s = 128; s > 0; s >>= 1) {
    if (tid < s) red[tid] += red[tid+s];
    __syncthreads();
  }
  if (tid == 0)
    out[0] = red[0] * (0.5f / (SIGMA_ * (float)(B_*P_*KS_)));
}

extern "C" void kernel_launch(void* const* d_in, const int* in_sizes, int n_in,
                              void* d_out, int out_size, void* d_ws, size_t ws_size,
                              hipStream_t stream)
{
  const float* noisy_pc   = (const float*)d_in[0];
  const float* clean_pc   = (const float*)d_in[1];
  const int*   sampled_idx= (const int*)  d_in[2];
  const float* Wf1 = (const float*)d_in[3];
  const float* bf1 = (const float*)d_in[4];
  const float* Wf2 = (const float*)d_in[5];
  const float* bf2 = (const float*)d_in[6];
  const float* Ws1 = (const float*)d_in[7];
  const float* bs1 = (const float*)d_in[8];
  const float* Ws2 = (const float*)d_in[9];
  const float* bs2 = (const float*)d_in[10];
  const float* Ws3 = (const float*)d_in[11];
  const float* bs3 = (const float*)d_in[12];

  float* ws = (float*)d_ws;
  float* f_origin   = ws;  ws += 512*3;
  float* noisy_feat = ws;  ws += 512*Z_;
  float* fgat       = ws;  ws += (long)B_*P_*KS_*3;
  float* estim      = ws;  ws += (long)B_*P_*KS_*3;
  float* ground     = ws;  ws += (long)B_*P_*KS_*3;

  k_feat <<<32,  32,  0, stream>>>(noisy_pc, sampled_idx, Wf1, bf1, Wf2, bf2, f_origin, noisy_feat);
  k_knn1 <<<B_,  128, 0, stream>>>(noisy_pc, f_origin, fgat);
  k_score<<<512, 64,  0, stream>>>(fgat, f_origin, noisy_feat, Ws1, bs1, Ws2, bs2, Ws3, bs3, estim);
  k_knn2 <<<64,  256, 0, stream>>>(clean_pc, fgat, ground);
  k_loss <<<1,   256, 0, stream>>>(estim, ground, (float*)d_out);
}